// Energy_65085934403613
// MI455X (gfx1250) — compile-verified
//
#include <hip/hip_runtime.h>
#include <stdint.h>

// Problem constants (fixed by the reference)
#define B_ 16
#define T_ 400
#define P_ 3500
#define S_ 10000
#define E_ 100000
#define NTHREADS 1024
#define CHUNKS ((P_ * 4) / 16)   // 875 x 16-byte async transfers per frame

__device__ __forceinline__ float wave_sum(float v) {
#pragma unroll
  for (int m = 16; m >= 1; m >>= 1) v += __shfl_xor(v, m, 32);
  return v;
}

// Pack edges into 16B records {src,dst,pdf,exp(logw)}, build 0.1*exp(init_logp),
// and zero the scalar output.
__global__ void __launch_bounds__(NTHREADS)
prep_kernel(const float* __restrict__ logw,
            const int* __restrict__ src,
            const int* __restrict__ dst,
            const int* __restrict__ pdf,
            const float* __restrict__ init_logp,
            uint4* __restrict__ edges,
            float* __restrict__ lk_init,
            float* __restrict__ out) {
  int i = blockIdx.x * blockDim.x + threadIdx.x;
  if (i == 0) out[0] = 0.0f;
  if (i < E_) {
    uint4 e;
    e.x = (unsigned)src[i];
    e.y = (unsigned)dst[i];
    e.z = (unsigned)pdf[i];
    e.w = __float_as_uint(__expf(logw[i]));
    edges[i] = e;
  }
  if (i < S_) lk_init[i] = 0.1f * __expf(init_logp[i]);
}

// One workgroup per sequence. Entire forward state machine lives in LDS:
//   A   : alpha-hat (normalized up to the deferred 1/Ssum factor), 40 KB
//   Nacc: scatter accumulator for next alpha, 40 KB
//   LK  : 0.1*exp(init_logp) leaky term, 40 KB
//   X   : double-buffered exp(x_t)*inv frame, 2 x 14 KB (async-staged)
// Total ~148 KB of the WGP's 320 KB LDS.
//
// Normalization trick: instead of sweeping A[s] *= 1/Ssum every step, the
// factor is folded into the next frame's exp pass (every edge contribution
// contains exactly one X value), keeping the sum(alpha-hat)==1 invariant the
// leaky term needs while deleting 2*S LDS ops + 1 barrier per timestep.
__global__ void __launch_bounds__(NTHREADS)
chain_kernel(const float* __restrict__ x,
             const float* __restrict__ final_logp,
             const uint4* __restrict__ edges,
             const float* __restrict__ lk_init,
             float* __restrict__ out) {
  __shared__ float A[S_];
  __shared__ float Nacc[S_];
  __shared__ float LK[S_];
  __shared__ float X[2][P_];
  __shared__ float red[33];

  const int tid  = threadIdx.x;
  const int b    = blockIdx.x;
  const int lane = tid & 31;
  const int wid  = tid >> 5;

  for (int s = tid; s < S_; s += NTHREADS) {
    float lk = lk_init[s];
    LK[s]   = lk;
    A[s]    = 10.0f * lk;   // exp(init_logp); sums to 1 (log_softmax input)
    Nacc[s] = 0.0f;
  }

  // Async-stage frame 0 into X[0] (global -> LDS, no VGPR round trip).
  if (tid < CHUNKS) {
    const float* g = x + ((size_t)b * T_) * P_;
    uint64_t gaddr = (uint64_t)(uintptr_t)(g + tid * 4);
    uint32_t laddr = (uint32_t)(uintptr_t)(&X[0][tid * 4]);
    asm volatile("global_load_async_to_lds_b128 %0, %1, off"
                 :: "v"(laddr), "v"(gaddr) : "memory");
  }
  __syncthreads();

  float c   = 0.0f;   // running log-normalizer (identical in every thread)
  float inv = 1.0f;   // deferred 1/Ssum from the previous timestep

  for (int t = 0; t < T_; ++t) {
    const int cur = t & 1, nxt = cur ^ 1;

    // Wait for this frame's async copy, then make it workgroup-visible.
    asm volatile("s_wait_asynccnt 0" ::: "memory");
    __syncthreads();

    // Kick off the async stage of frame t+1 into the other buffer (free since
    // the end of the previous edge pass); overlaps exp pass + edge pass.
    if (t + 1 < T_ && tid < CHUNKS) {
      const float* g = x + ((size_t)b * T_ + (t + 1)) * P_;
      uint64_t gaddr = (uint64_t)(uintptr_t)(g + tid * 4);
      uint32_t laddr = (uint32_t)(uintptr_t)(&X[nxt][tid * 4]);
      asm volatile("global_load_async_to_lds_b128 %0, %1, off"
                   :: "v"(laddr), "v"(gaddr) : "memory");
    }

    // exp() the current frame in place, folding in the deferred normalizer.
    for (int p = tid; p < P_; p += NTHREADS)
      X[cur][p] = __expf(X[cur][p]) * inv;
    __syncthreads();

    // Hot loop: one b128 load from L2-resident packed edges, two LDS gathers,
    // two fmuls, one ds_add_f32 scatter. ~98 edges per thread.
#pragma unroll 2
    for (int e = tid; e < E_; e += NTHREADS) {
      uint4 ed = edges[e];
      float v = A[ed.x] * __uint_as_float(ed.w) * X[cur][ed.z];
      atomicAdd(&Nacc[ed.y], v);
    }
    __syncthreads();

    // alpha' = N + leaky; accumulate Ssum; defer 1/Ssum to next exp pass.
    float loc = 0.0f;
    for (int s = tid; s < S_; s += NTHREADS) {
      float v = Nacc[s] + LK[s];
      A[s]    = v;
      Nacc[s] = 0.0f;        // ready for next timestep
      loc    += v;
    }
    loc = wave_sum(loc);
    if (lane == 0) red[wid] = loc;
    __syncthreads();
    if (wid == 0) {
      float w = wave_sum(red[lane]);
      if (lane == 0) red[32] = w;
    }
    __syncthreads();
    float Ssum = red[32];
    c  += __logf(Ssum);
    inv = 1.0f / Ssum;
  }

  // logprob = c + log( inv * sum_s A[s] * exp(final_logp[s]) );  out -= logprob.
  float loc = 0.0f;
  for (int s = tid; s < S_; s += NTHREADS)
    loc += A[s] * __expf(final_logp[s]);
  loc = wave_sum(loc);
  if (lane == 0) red[wid] = loc;
  __syncthreads();
  if (wid == 0) {
    float w = wave_sum(red[lane]);
    if (lane == 0) red[32] = w;
  }
  __syncthreads();
  if (tid == 0) atomicAdd(out, -(c + __logf(inv * red[32])));
}

extern "C" void kernel_launch(void* const* d_in, const int* in_sizes, int n_in,
                              void* d_out, int out_size, void* d_ws, size_t ws_size,
                              hipStream_t stream) {
  (void)in_sizes; (void)n_in; (void)out_size; (void)ws_size;
  const float* x          = (const float*)d_in[0];
  const float* edge_logw  = (const float*)d_in[1];
  const float* init_logp  = (const float*)d_in[2];
  const float* final_logp = (const float*)d_in[3];
  const int*   edge_src   = (const int*)d_in[4];
  const int*   edge_dst   = (const int*)d_in[5];
  const int*   edge_pdf   = (const int*)d_in[6];
  float* out = (float*)d_out;

  // Scratch layout: [packed edges: E*16B][lk_init: S*4B]  (~1.64 MB)
  uint4* edges   = (uint4*)d_ws;
  float* lk_init = (float*)((char*)d_ws + (size_t)E_ * sizeof(uint4));

  int prep_blocks = (E_ + NTHREADS - 1) / NTHREADS;
  prep_kernel<<<prep_blocks, NTHREADS, 0, stream>>>(
      edge_logw, edge_src, edge_dst, edge_pdf, init_logp, edges, lk_init, out);
  chain_kernel<<<B_, NTHREADS, 0, stream>>>(x, final_logp, edges, lk_init, out);
}